// TensorNeRF_25065429139671
// MI455X (gfx1250) — compile-verified
//
#include <hip/hip_runtime.h>
#include <hip/hip_bf16.h>
#include <stdint.h>

// NeRF volume-rendering composite: R rays x S samples, wave-per-ray.
// Memory bound (~352 MB streamed once) -> NT b128 loads, async global->LDS
// staging of the rgb stream (ASYNCcnt path), wave32 product scan, local-prefix
// factorized weighting, exp2/log2-fused alpha math, 32-bit addressing.

#define R_COUNT   65536
#define S_COUNT   256
#define LANES     32
#define CHUNK     (S_COUNT / LANES)   // 8 samples per lane
#define WPB       8                   // waves (rays) per block
#define BLOCK_THREADS (WPB * LANES)   // 256
#define LOG2E     1.44269504088896340736f

typedef float v4f __attribute__((ext_vector_type(4)));
typedef int   v4i __attribute__((ext_vector_type(4)));

#if defined(__AMDGCN__) && defined(__gfx1250__) && \
    __has_builtin(__builtin_amdgcn_global_load_async_to_lds_b128)
#define USE_ASYNC 1
#else
#define USE_ASYNC 0
#endif

__global__ __launch_bounds__(BLOCK_THREADS) void nerf_vr_kernel(
    const float* __restrict__ sigma_feature,     // (R,S)
    const float* __restrict__ rgb,               // (R,S,3)
    const float* __restrict__ z_vals,            // (R,S)
    const unsigned char* __restrict__ ray_valid, // (R,S) bool bytes
    float* __restrict__ out)                     // [3R rgb | R depth | R acc | R bg]
{
    const int lane = threadIdx.x & (LANES - 1);
    const int wv   = threadIdx.x >> 5;
    const int ray  = blockIdx.x * WPB + wv;

    // All arrays < 2^31 bytes -> 32-bit offsets (saddr + voffset addressing).
    const unsigned boff = (unsigned)ray * S_COUNT + (unsigned)(lane * CHUNK);

#if USE_ASYNC
    __shared__ float ldsRGB[WPB * S_COUNT * 3];  // 24 KB / block
    {
        typedef __attribute__((address_space(1))) char gch;
        typedef __attribute__((address_space(3))) char lch;
        typedef __attribute__((address_space(1))) v4i gvec;
        typedef __attribute__((address_space(3))) v4i lvec;
        gch* gsrc = (gch*)(void*)(rgb + (unsigned)ray * (S_COUNT * 3));
        lch* ldst = (lch*)(void*)(&ldsRGB[wv * S_COUNT * 3]);
        // 6 x (32 lanes x 16B) = 3072 B = this ray's rgb chunk
        #pragma unroll
        for (int i = 0; i < 6; ++i) {
            __builtin_amdgcn_global_load_async_to_lds_b128(
                (gvec*)(gsrc + i * 512 + lane * 16),
                (lvec*)(ldst + i * 512 + lane * 16),
                0, 0);
        }
    }
#endif

    // Streaming inputs: touched once, 352 MB > 192 MB L2 -> non-temporal.
    const v4f* sp = (const v4f*)(sigma_feature + boff);
    const v4f* zp = (const v4f*)(z_vals + boff);
    v4f s0 = __builtin_nontemporal_load(sp);
    v4f s1 = __builtin_nontemporal_load(sp + 1);
    v4f z0 = __builtin_nontemporal_load(zp);
    v4f z1 = __builtin_nontemporal_load(zp + 1);
    unsigned long long vm =
        __builtin_nontemporal_load((const unsigned long long*)(ray_valid + boff));
    const unsigned vlo = (unsigned)vm;
    const unsigned vhi = (unsigned)(vm >> 32);

    float zz[CHUNK], wj[CHUNK];
    #pragma unroll
    for (int j = 0; j < 4; ++j) { zz[j] = z0[j]; zz[4 + j] = z1[j]; }

    // Next lane's first z, for the dist at our chunk boundary.
    float zn = __shfl_down(zz[0], 1, LANES);

    // alpha = valid * (1 - (1+e^x)^(-25 d)),  x = sigma - 10
    //       = valid * (1 - exp2(-25 d * log2(1 + exp2(x*log2e))))
    // Accumulate with the LOCAL prefix Tl; rescale by the wave-scan exclusive
    // prefix afterwards (w_j = alpha_j * T_excl * Tl_j).
    float Tl = 1.0f;
    float aW = 0.0f, aD = 0.0f;
    #pragma unroll
    for (int j = 0; j < CHUNK; ++j) {
        float d  = (j < CHUNK - 1)
                       ? (zz[j + 1] - zz[j])
                       : ((lane == LANES - 1) ? 100.0f : (zn - zz[CHUNK - 1]));
        float sg = (j < 4) ? s0[j] : s1[j - 4];
        float e  = __builtin_amdgcn_exp2f(__builtin_fmaf(sg, LOG2E, -10.0f * LOG2E));
        float u  = 1.0f + e;                        // 1 + e^x  (stable)
        float lg = __builtin_amdgcn_logf(u);        // v_log_f32 = log2(u)
        float ap = __builtin_amdgcn_exp2f((d * lg) * -25.0f); // exp(-softplus*25*d)
        float vf = (float)((((j < 4) ? vlo : vhi) >> (8 * (j & 3))) & 0xffu); // 0/1
        float a  = __builtin_fmaf(-vf, ap, vf);     // alpha = vf*(1-ap)
        float t  = (1.0f - a) + 1e-10f;
        float w  = a * Tl;                          // local-prefix weight
        wj[j] = w;
        aW += w;
        aD  = __builtin_fmaf(w, zz[j], aD);
        Tl *= t;
    }

    // Wave32 inclusive product scan of per-lane products (transmittance prefix).
    float incl = Tl;
    #pragma unroll
    for (int off = 1; off < LANES; off <<= 1) {
        float t = __shfl_up(incl, off, LANES);
        if (lane >= off) incl *= t;
    }
    float bg = __shfl(incl, LANES - 1, LANES);      // T over all 256 samples
    float T  = __shfl_up(incl, 1, LANES);           // exclusive prefix
    if (lane == 0) T = 1.0f;

    // rgb chunk: 6 x v4f = 24 floats (8 samples x rgb)
    v4f c[6];
#if USE_ASYNC
  #if __has_builtin(__builtin_amdgcn_s_wait_asynccnt)
    __builtin_amdgcn_s_wait_asynccnt(0);
  #else
    asm volatile("s_wait_asynccnt 0x0" ::: "memory");
  #endif
    asm volatile("" ::: "memory");
    {
        const v4f* rp4 = (const v4f*)&ldsRGB[(wv * S_COUNT + lane * CHUNK) * 3];
        #pragma unroll
        for (int i = 0; i < 6; ++i) c[i] = rp4[i];   // ds_load_b128 x6
    }
#else
    {
        const v4f* cp = (const v4f*)(rgb + (unsigned)(boff * 3)); // 16B aligned
        #pragma unroll
        for (int i = 0; i < 6; ++i) c[i] = __builtin_nontemporal_load(cp + i);
    }
#endif

    float aR = 0.0f, aG = 0.0f, aB = 0.0f;
    #pragma unroll
    for (int j = 0; j < CHUNK; ++j) {
        const int k = 3 * j;
        float w = wj[j];
        aR = __builtin_fmaf(w, c[(k + 0) >> 2][(k + 0) & 3], aR);
        aG = __builtin_fmaf(w, c[(k + 1) >> 2][(k + 1) & 3], aG);
        aB = __builtin_fmaf(w, c[(k + 2) >> 2][(k + 2) & 3], aB);
    }

    // Per-lane rescale by the exclusive wave prefix, then butterfly reduce.
    aW *= T; aD *= T; aR *= T; aG *= T; aB *= T;
    #pragma unroll
    for (int off = LANES / 2; off > 0; off >>= 1) {
        aW += __shfl_xor(aW, off, LANES);
        aD += __shfl_xor(aD, off, LANES);
        aR += __shfl_xor(aR, off, LANES);
        aG += __shfl_xor(aG, off, LANES);
        aB += __shfl_xor(aB, off, LANES);
    }

    if (lane == 0) {
        __builtin_nontemporal_store(aR, out + (unsigned)ray * 3 + 0);
        __builtin_nontemporal_store(aG, out + (unsigned)ray * 3 + 1);
        __builtin_nontemporal_store(aB, out + (unsigned)ray * 3 + 2);
        __builtin_nontemporal_store(aD, out + 3u * R_COUNT + (unsigned)ray);
        __builtin_nontemporal_store(aW, out + 4u * R_COUNT + (unsigned)ray);
        __builtin_nontemporal_store(bg, out + 5u * R_COUNT + (unsigned)ray);
    }
}

extern "C" void kernel_launch(void* const* d_in, const int* in_sizes, int n_in,
                              void* d_out, int out_size, void* d_ws, size_t ws_size,
                              hipStream_t stream) {
    const float*         sigma = (const float*)d_in[0];
    const float*         rgb   = (const float*)d_in[1];
    const float*         z     = (const float*)d_in[2];
    const unsigned char* valid = (const unsigned char*)d_in[3];
    float*               out   = (float*)d_out;

    dim3 grid(R_COUNT / WPB);
    dim3 block(BLOCK_THREADS);
    nerf_vr_kernel<<<grid, block, 0, stream>>>(sigma, rgb, z, valid, out);
}